// WarpMesh_38439957299632
// MI455X (gfx1250) — compile-verified
//
#include <hip/hip_runtime.h>
#include <hip/hip_bf16.h>

typedef __attribute__((ext_vector_type(2))) float v2f;
typedef __attribute__((ext_vector_type(8))) float v8f;

#define HH 768
#define WW 768
#define NFACES 128
#define NGROUPS 32                 // 128 faces / 4 faces per WMMA
#define COEF_FLOATS (NFACES * 16)  // 4 quantities x [cx, cy, c0, 0] per face

// ---------------------------------------------------------------------------
// Kernel 1: per-face affine coefficients.
// For face f, quantity q in {wb, wc, sx, sy}:  val(px,py) = cx*px + cy*py + c0
// Layout: coef[f*16 + q*4 + {0:cx, 1:cy, 2:c0, 3:0}]
// ---------------------------------------------------------------------------
__global__ void WarpMesh_coef_kernel(const float* __restrict__ points,
                                     const float* __restrict__ uv,
                                     const int* __restrict__ faces,
                                     float* __restrict__ coef) {
    int f = threadIdx.x;
    if (f >= NFACES) return;

    int i0 = faces[3 * f + 0], i1 = faces[3 * f + 1], i2 = faces[3 * f + 2];
    bool skip = (i0 == 0) && (i1 == 0) && (i2 == 0);

    const float kx = 2.0f / (float)(WW - 1);
    const float ky = 2.0f / (float)(HH - 1);

    // normalized triangle vertices
    float ax = points[2 * i0] * kx - 1.0f, ay = points[2 * i0 + 1] * ky - 1.0f;
    float bx = points[2 * i1] * kx - 1.0f, by = points[2 * i1 + 1] * ky - 1.0f;
    float cx = points[2 * i2] * kx - 1.0f, cy = points[2 * i2 + 1] * ky - 1.0f;

    float v0x = bx - ax, v0y = by - ay;
    float v1x = cx - ax, v1y = cy - ay;
    float d00 = v0x * v0x + v0y * v0y;
    float d01 = v0x * v1x + v0y * v1y;
    float d11 = v1x * v1x + v1y * v1y;
    float denom = d00 * d11 - d01 * d01;

    bool bad = skip || !(fabsf(denom) > 1e-30f) || !(fabsf(denom) < 1e30f);
    float inv = bad ? 0.0f : 1.0f / denom;

    // wb = u . (coords - a), wc = w . (coords - a)   (normalized coords)
    float ux = (d11 * v0x - d01 * v1x) * inv, uy = (d11 * v0y - d01 * v1y) * inv;
    float wx = (d00 * v1x - d01 * v0x) * inv, wy = (d00 * v1y - d01 * v0y) * inv;

    // convert to pixel-coordinate affine form: gx = kx*px - 1, gy = ky*py - 1
    float cbx = ux * kx, cby = uy * ky, cb0 = -ux - uy - (ux * ax + uy * ay);
    float ccx = wx * kx, ccy = wy * ky, cc0 = -wx - wy - (wx * ax + wy * ay);
    if (bad) { cbx = 0.f; cby = 0.f; cb0 = -1e30f;   // wb always < -eps
               ccx = 0.f; ccy = 0.f; cc0 = -1e30f; }

    // sample coords: since wa+wb+wc==1, de-normalized grid_sample coords are the
    // direct pixel-space UV interpolation: s = uvA + (uvB-uvA)*wb + (uvC-uvA)*wc
    float uax = uv[2 * i0], uay = uv[2 * i0 + 1];
    float ubx = uv[2 * i1], uby = uv[2 * i1 + 1];
    float ucx = uv[2 * i2], ucy = uv[2 * i2 + 1];
    float dbx = ubx - uax, dcx = ucx - uax;
    float dby = uby - uay, dcy = ucy - uay;

    float sxx = dbx * cbx + dcx * ccx;
    float sxy = dbx * cby + dcx * ccy;
    float sx0 = uax + dbx * cb0 + dcx * cc0;
    float syx = dby * cbx + dcy * ccx;
    float syy = dby * cby + dcy * ccy;
    float sy0 = uay + dby * cb0 + dcy * cc0;

    float* o = coef + f * 16;
    o[0]  = cbx; o[1]  = cby; o[2]  = cb0; o[3]  = 0.0f;  // wb
    o[4]  = ccx; o[5]  = ccy; o[6]  = cc0; o[7]  = 0.0f;  // wc
    o[8]  = sxx; o[9]  = sxy; o[10] = sx0; o[11] = 0.0f;  // sx
    o[12] = syx; o[13] = syy; o[14] = sy0; o[15] = 0.0f;  // sy
}

// ---------------------------------------------------------------------------
// Kernel 2: WMMA rasterization. 256 threads = 8 waves; each wave owns a
// 16-pixel horizontal strip. Per WMMA: A = 16 rows (4 faces x {wb,wc,sx,sy}),
// K=4 coefficient basis; B = 4 x 16 pixels [px, py, 1, 0].
// D layout gives each lane one pixel: lanes 0-15 see faces 4k,4k+1 (rows 0-7),
// lanes 16-31 see faces 4k+2,4k+3 (rows 8-15). Priority merge is lane-local
// plus one half-wave shuffle at the end.
// ---------------------------------------------------------------------------
__global__ void __launch_bounds__(256)
WarpMesh_warp_kernel(const float* __restrict__ image,
                     const float* __restrict__ coef,
                     float* __restrict__ out) {
    __shared__ float lds[COEF_FLOATS];
    for (int i = threadIdx.x; i < COEF_FLOATS; i += 256) lds[i] = coef[i];
    __syncthreads();

    const int lane = threadIdx.x & 31;
    const int wave = threadIdx.x >> 5;
    const int col  = lane & 15;
    const bool hi  = lane >= 16;

    const int tile  = blockIdx.x * 8 + wave;
    const int pbase = tile * 16;          // 48 tiles per row; never straddles
    const int py    = pbase / WW;
    const int px    = (pbase % WW) + col;

    // B: lanes 0-15 -> (K0,K1)=(px,py); lanes 16-31 -> (K2,K3)=(1,0)
    const v2f bvec = hi ? (v2f){1.0f, 0.0f} : (v2f){(float)px, (float)py};

    const float eps = 1.0f / 767.0f;
    int   bestF  = -1;
    float bestSx = 0.0f, bestSy = 0.0f;

    for (int k = 0; k < NGROUPS; ++k) {
        // A: row m = col, lanes 0-15 load (cx,cy), lanes 16-31 load (c0,0)
        const v2f avec = *(const v2f*)&lds[(k * 16 + col) * 4 + (hi ? 2 : 0)];
        v8f acc = {};
        acc = __builtin_amdgcn_wmma_f32_16x16x4_f32(
            /*neg_a=*/false, avec, /*neg_b=*/false, bvec,
            /*c_mod=*/(short)0, acc, /*reuse_a=*/false, /*reuse_b=*/false);

        const int fA = 4 * k + (hi ? 2 : 0);
        {   // first local face: rows +0..3 -> wb, wc, sx, sy
            float wb = acc[0], wc = acc[1];
            bool valid = (wb >= -eps) && (wb <= 1.0f + eps) &&
                         (wc >= -eps) && (wc <= 1.0f + eps) &&
                         (wb + wc <= 1.0f + eps);
            if (valid) { bestF = fA; bestSx = acc[2]; bestSy = acc[3]; }
        }
        {   // second local face: rows +4..7 (higher index, overwrites)
            float wb = acc[4], wc = acc[5];
            bool valid = (wb >= -eps) && (wb <= 1.0f + eps) &&
                         (wc >= -eps) && (wc <= 1.0f + eps) &&
                         (wb + wc <= 1.0f + eps);
            if (valid) { bestF = fA + 1; bestSx = acc[6]; bestSy = acc[7]; }
        }
    }

    // merge across half-waves: partner holds the other two faces mod 4
    int   pF  = __shfl_xor(bestF,  16, 32);
    float pSx = __shfl_xor(bestSx, 16, 32);
    float pSy = __shfl_xor(bestSy, 16, 32);
    if (pF > bestF) { bestF = pF; bestSx = pSx; bestSy = pSy; }

    // bilinear sample (align_corners=True, zeros padding per tap)
    float r0, r1, r2;
    if (bestF < 0) {
        r0 = r1 = r2 = 1.0f;                    // jnp.ones_like init
    } else {
        float x  = bestSx, y = bestSy;
        float x0 = floorf(x), y0 = floorf(y);
        float wx1 = x - x0, wy1 = y - y0;
        r0 = r1 = r2 = 0.0f;
#pragma unroll
        for (int t = 0; t < 4; ++t) {
            float xi = x0 + (float)(t & 1);
            float yi = y0 + (float)(t >> 1);
            float w  = ((t & 1) ? wx1 : 1.0f - wx1) *
                       ((t >> 1) ? wy1 : 1.0f - wy1);
            bool v = (xi >= 0.0f) && (xi <= (float)(WW - 1)) &&
                     (yi >= 0.0f) && (yi <= (float)(HH - 1));
            int xic = (int)fminf(fmaxf(xi, 0.0f), (float)(WW - 1));
            int yic = (int)fminf(fmaxf(yi, 0.0f), (float)(HH - 1));
            const float* p = image + (yic * WW + xic) * 3;
            float wv = v ? w : 0.0f;
            r0 += p[0] * wv; r1 += p[1] * wv; r2 += p[2] * wv;
        }
    }

    // both half-lanes hold the same pixel: split the 3-channel store
    float* o = out + (pbase + col) * 3;
    if (!hi) { o[0] = r0; o[1] = r1; }
    else     { o[2] = r2; }
}

// ---------------------------------------------------------------------------
extern "C" void kernel_launch(void* const* d_in, const int* in_sizes, int n_in,
                              void* d_out, int out_size, void* d_ws, size_t ws_size,
                              hipStream_t stream) {
    const float* image  = (const float*)d_in[0];   // (768,768,3) f32
    const float* points = (const float*)d_in[1];   // (81,2) f32
    const float* uv     = (const float*)d_in[2];   // (81,2) f32
    const int*   faces  = (const int*)d_in[3];     // (128,3) i32
    (void)in_sizes; (void)n_in; (void)out_size; (void)ws_size;

    float* coef = (float*)d_ws;                    // 8 KB scratch

    WarpMesh_coef_kernel<<<1, 128, 0, stream>>>(points, uv, faces, coef);

    const int tiles  = (HH * WW) / 16;             // 36864
    const int blocks = tiles / 8;                  // 4608 blocks x 8 waves
    WarpMesh_warp_kernel<<<blocks, 256, 0, stream>>>(image, coef, (float*)d_out);
}